// ParallelBranchScaledLayerHackRMSNorm_78073915506894
// MI455X (gfx1250) — compile-verified
//
#include <hip/hip_runtime.h>
#include <hip/hip_bf16.h>

typedef __attribute__((ext_vector_type(2))) float v2f;
typedef __attribute__((ext_vector_type(4))) float v4f;
typedef __attribute__((ext_vector_type(8))) float v8f;

#define MDIM 16384   // B*S
#define NDIM 1024    // H (output)
#define KDIM 1024    // H (input)
#define NLAYERS 20
#define EPSV 1e-6f

// ---------------------------------------------------------------------------
// Kernel 1: w_sum = sum over 20 layers of conv_weights  (80MB -> 4MB, mem-bound)
// ---------------------------------------------------------------------------
__global__ __launch_bounds__(256) void wsum_kernel(const float* __restrict__ cw,
                                                   float* __restrict__ ws) {
    size_t idx = ((size_t)blockIdx.x * blockDim.x + threadIdx.x) * 4;  // float4 granularity
    v4f acc = *(const v4f*)(cw + idx);
#pragma unroll
    for (int l = 1; l < NLAYERS; ++l) {
        acc += *(const v4f*)(cw + (size_t)l * (NDIM * KDIM) + idx);
    }
    *(v4f*)(ws + idx) = acc;
}

// ---------------------------------------------------------------------------
// Kernel 2: scaled-layerhack RMSNorm. mean(concat(xs,-xs)) == 0 exactly, so
//   out = xs * rsqrt(mean(xs^2) + eps) * w,  xs = x / max(|x|, 1e-6)
// One 256-thread block per row (1024 elems, 4 per thread). Memory-bound.
// ---------------------------------------------------------------------------
__global__ __launch_bounds__(256) void rmsnorm_kernel(const float* __restrict__ x,
                                                      const float* __restrict__ w,
                                                      float* __restrict__ out) {
    const int row  = blockIdx.x;
    const int tid  = threadIdx.x;
    const int lane = tid & 31;
    const int wid  = tid >> 5;
    const float* xr = x + (size_t)row * 1024;
    float* orow     = out + (size_t)row * 1024;

    __shared__ float sm[8];

    v4f v = *(const v4f*)(xr + tid * 4);
    float mx = fmaxf(fmaxf(fabsf(v.x), fabsf(v.y)), fmaxf(fabsf(v.z), fabsf(v.w)));
#pragma unroll
    for (int off = 16; off > 0; off >>= 1)
        mx = fmaxf(mx, __shfl_xor(mx, off, 32));
    if (lane == 0) sm[wid] = mx;
    __syncthreads();
    if (tid < 32) {
        float t = (tid < 8) ? sm[tid] : 0.0f;
#pragma unroll
        for (int off = 4; off > 0; off >>= 1)
            t = fmaxf(t, __shfl_xor(t, off, 32));
        if (tid == 0) sm[0] = fmaxf(t, 1e-6f);
    }
    __syncthreads();
    const float inv = 1.0f / sm[0];

    v4f xs = v * inv;
    float ss = xs.x * xs.x + xs.y * xs.y + xs.z * xs.z + xs.w * xs.w;
    __syncthreads();   // done reading sm[0]; safe to reuse sm
#pragma unroll
    for (int off = 16; off > 0; off >>= 1)
        ss += __shfl_xor(ss, off, 32);
    if (lane == 0) sm[wid] = ss;
    __syncthreads();
    if (tid < 32) {
        float t = (tid < 8) ? sm[tid] : 0.0f;
#pragma unroll
        for (int off = 4; off > 0; off >>= 1)
            t += __shfl_xor(t, off, 32);
        if (tid == 0) sm[0] = rsqrtf(t * (1.0f / 1024.0f) + EPSV);
    }
    __syncthreads();
    const float rstd = sm[0];

    v4f wv = *(const v4f*)(w + tid * 4);
    v4f o  = xs * rstd * wv;
    *(v4f*)(orow + tid * 4) = o;
}

// ---------------------------------------------------------------------------
// Kernel 3: C = (X . w_sum^T) / 20 via V_WMMA_F32_16X16X4_F32 (full fp32).
// Block = 256 thr (8 waves) -> 128x64 C tile; wave = 32x32 (4 accumulators).
// Per k-step-of-4 a lane reads 4 v2f from LDS and issues 4 WMMAs: both A and
// B fragments reused 2x (8 FLOP / LDS byte), depth-4 accumulator interleave.
// LDS rows padded to 36 floats (16B-aligned, conflict-free column reads).
// ---------------------------------------------------------------------------
#define TM 128
#define TN 64
#define TK 32
#define LDSW 36   // padded LDS row stride in floats (36*4 = 144B)

__global__ __launch_bounds__(256) void gemm_wmma_kernel(const float* __restrict__ X,
                                                        const float* __restrict__ Wm,
                                                        float* __restrict__ C) {
    __shared__ float As[TM * LDSW];   // 128 x 32 tile of X
    __shared__ float Bs[TN * LDSW];   //  64 x 32 tile of w_sum (row n = output col)

    const int tid  = threadIdx.x;
    const int lane = tid & 31;
    const int wid  = tid >> 5;     // 0..7
    const int wm   = wid & 3;      // wave row group: 32 rows
    const int wn   = wid >> 2;     // wave col group: 32 cols
    const int mblk = blockIdx.y * TM;
    const int nblk = blockIdx.x * TN;

    v8f acc00 = {};   // (rows 0-15 , cols 0-15 ) of wave tile
    v8f acc01 = {};   // (rows 0-15 , cols 16-31)
    v8f acc10 = {};   // (rows 16-31, cols 0-15 )
    v8f acc11 = {};   // (rows 16-31, cols 16-31)

    // staging map (float4 granularity): row = f>>3, col = (f&7)*4
    const int srow = tid >> 3;          // 0..31
    const int scol = (tid & 7) << 2;    // 0,4,...,28

    // fragment addresses
    const int arow0 = wm * 32 + (lane & 15);
    const int arow1 = arow0 + 16;
    const int ncol0 = wn * 32 + (lane & 15);
    const int ncol1 = ncol0 + 16;
    const int kb    = (lane >> 4) << 1;   // K sub-offset 0 or 2

    for (int k0 = 0; k0 < KDIM; k0 += TK) {
        // A: 128x32 = 1024 float4, 4 per thread; B: 64x32 = 512 float4, 2 per thread
        v4f a0 = *(const v4f*)(X  + (size_t)(mblk + srow      ) * KDIM + k0 + scol);
        v4f a1 = *(const v4f*)(X  + (size_t)(mblk + srow + 32 ) * KDIM + k0 + scol);
        v4f a2 = *(const v4f*)(X  + (size_t)(mblk + srow + 64 ) * KDIM + k0 + scol);
        v4f a3 = *(const v4f*)(X  + (size_t)(mblk + srow + 96 ) * KDIM + k0 + scol);
        v4f b0 = *(const v4f*)(Wm + (size_t)(nblk + srow      ) * KDIM + k0 + scol);
        v4f b1 = *(const v4f*)(Wm + (size_t)(nblk + srow + 32 ) * KDIM + k0 + scol);
        __syncthreads();
        *(v4f*)(As + (srow      ) * LDSW + scol) = a0;
        *(v4f*)(As + (srow + 32 ) * LDSW + scol) = a1;
        *(v4f*)(As + (srow + 64 ) * LDSW + scol) = a2;
        *(v4f*)(As + (srow + 96 ) * LDSW + scol) = a3;
        *(v4f*)(Bs + (srow      ) * LDSW + scol) = b0;
        *(v4f*)(Bs + (srow + 32 ) * LDSW + scol) = b1;
        __syncthreads();

#pragma unroll
        for (int kk = 0; kk < TK; kk += 4) {
            v2f af0 = *(const v2f*)(As + arow0 * LDSW + kk + kb);
            v2f af1 = *(const v2f*)(As + arow1 * LDSW + kk + kb);
            v2f bf0 = *(const v2f*)(Bs + ncol0 * LDSW + kk + kb);
            v2f bf1 = *(const v2f*)(Bs + ncol1 * LDSW + kk + kb);
            acc00 = __builtin_amdgcn_wmma_f32_16x16x4_f32(
                false, af0, false, bf0, (short)0, acc00, false, false);
            acc01 = __builtin_amdgcn_wmma_f32_16x16x4_f32(
                false, af0, false, bf1, (short)0, acc01, false, false);
            acc10 = __builtin_amdgcn_wmma_f32_16x16x4_f32(
                false, af1, false, bf0, (short)0, acc10, false, false);
            acc11 = __builtin_amdgcn_wmma_f32_16x16x4_f32(
                false, af1, false, bf1, (short)0, acc11, false, false);
        }
    }

    // C/D layout: VGPR i -> M = i (lanes 0-15) / M = i+8 (lanes 16-31); N = lane%16
    const float s  = 1.0f / (float)NLAYERS;
    const int row0 = mblk + wm * 32 + (lane >> 4) * 8;
    const int col0 = nblk + wn * 32 + (lane & 15);
#pragma unroll
    for (int i = 0; i < 8; ++i) {
        C[(size_t)(row0 + i     ) * NDIM + col0     ] = acc00[i] * s;
        C[(size_t)(row0 + i     ) * NDIM + col0 + 16] = acc01[i] * s;
        C[(size_t)(row0 + i + 16) * NDIM + col0     ] = acc10[i] * s;
        C[(size_t)(row0 + i + 16) * NDIM + col0 + 16] = acc11[i] * s;
    }
}

// ---------------------------------------------------------------------------
extern "C" void kernel_launch(void* const* d_in, const int* in_sizes, int n_in,
                              void* d_out, int out_size, void* d_ws, size_t ws_size,
                              hipStream_t stream) {
    const float* x  = (const float*)d_in[0];   // (4,4096,1024) f32
    const float* cw = (const float*)d_in[1];   // (20,1024,1024) f32
    const float* nw = (const float*)d_in[2];   // (1024,) f32

    float* rms_out  = (float*)d_out;                       // first output
    float* conv_out = rms_out + (size_t)MDIM * NDIM;       // second output
    float* wsum     = (float*)d_ws;                        // 4 MB scratch

    // 1) reduce 20 layers -> w_sum (1024x1024)
    wsum_kernel<<<(NDIM * KDIM) / 4 / 256, 256, 0, stream>>>(cw, wsum);

    // 2) rmsnorm branch (independent of w_sum)
    rmsnorm_kernel<<<MDIM, 256, 0, stream>>>(x, nw, rms_out);

    // 3) fp32 WMMA GEMM: C = X . w_sum^T / 20
    dim3 grid(NDIM / TN, MDIM / TM);
    gemm_wmma_kernel<<<grid, 256, 0, stream>>>(x, wsum, conv_out);
}